// SECOND_BEAM_SEARCH_37391985279367
// MI455X (gfx1250) — compile-verified
//
#include <hip/hip_runtime.h>
#include <cstdint>

#ifndef INF_F
#define INF_F __builtin_inff()
#endif

// ---- problem constants -----------------------------------------------------
static constexpr int    L = 28, B = 3, H = 16, S = 1024, D = 128;
static constexpr int    VOCAB = 151936;
static constexpr int    HIST = 64;
static constexpr int    NCHUNK = 32;                 // vocab chunks per batch
static constexpr int    CH = VOCAB / NCHUNK;         // 4748 (exact)
static_assert(VOCAB % NCHUNK == 0, "chunking must be exact");

static constexpr size_t TILE_FLOATS = (size_t)H * S * D;           // 2,097,152
static constexpr size_t KV_FLOATS   = (size_t)L * B * TILE_FLOATS; // 176,160,768

// output layout (flat, fp32)
static constexpr size_t OFF_KV     = 0;
static constexpr size_t OFF_TOK    = KV_FLOATS;                  // 3
static constexpr size_t OFF_SAVE   = OFF_TOK + 3;                // 3*65
static constexpr size_t OFF_RP     = OFF_SAVE + 3 * (HIST + 1);  // 3*VOCAB
static constexpr size_t OFF_PROB   = OFF_RP + (size_t)3 * VOCAB; // 3
static constexpr size_t OFF_MAXIDX = OFF_PROB + 3;               // 1

// workspace layout (bytes)
static constexpr size_t WS_BEAM = 0;     // int[3]
static constexpr size_t WS_TOK  = 16;    // int[3]
static constexpr size_t WS_PM   = 64;    // float[96]
static constexpr size_t WS_PS   = 448;   // float[96]
static constexpr size_t WS_PTV  = 832;   // float[288]
static constexpr size_t WS_PTI  = 1984;  // int[288]

// stable top-3 insert: order by (value desc, index asc) to match jax top_k
__device__ __forceinline__ void insert3(float xv, int xi, float* bv, int* bi) {
    if (xi < 0) return;
    bool g0 = (xv > bv[0]) || (xv == bv[0] && xi < bi[0]);
    bool g1 = (xv > bv[1]) || (xv == bv[1] && xi < bi[1]);
    bool g2 = (xv > bv[2]) || (xv == bv[2] && xi < bi[2]);
    if (g0)      { bv[2]=bv[1]; bi[2]=bi[1]; bv[1]=bv[0]; bi[1]=bi[0]; bv[0]=xv; bi[0]=xi; }
    else if (g1) { bv[2]=bv[1]; bi[2]=bi[1]; bv[1]=xv; bi[1]=xi; }
    else if (g2) { bv[2]=xv; bi[2]=xi; }
}

// ---- K1: per (batch, vocab-chunk) partial logsumexp + top-3 ----------------
__global__ __launch_bounds__(256)
void partial_topk(const float* __restrict__ logits, const float* __restrict__ rp,
                  float* __restrict__ pm, float* __restrict__ ps,
                  float* __restrict__ ptv, int* __restrict__ pti) {
    const int b  = blockIdx.x / NCHUNK;
    const int c  = blockIdx.x % NCHUNK;
    const int v0 = c * CH;
    const float* lrow = logits + (size_t)b * VOCAB;
    const float* rrow = rp     + (size_t)b * VOCAB;

    float m = -INF_F, s = 0.f;
    float tv0 = -INF_F, tv1 = -INF_F, tv2 = -INF_F;
    int   ti0 = -1, ti1 = -1, ti2 = -1;

    for (int v = v0 + (int)threadIdx.x; v < v0 + CH; v += 256) {
        float x = lrow[v] * rrow[v];
        if (x > m) { s = s * expf(m - x) + 1.f; m = x; }
        else       { s += expf(x - m); }
        // ascending v scan + strict > keeps lowest index on ties (stable)
        if (x > tv0)      { tv2=tv1; ti2=ti1; tv1=tv0; ti1=ti0; tv0=x; ti0=v; }
        else if (x > tv1) { tv2=tv1; ti2=ti1; tv1=x; ti1=v; }
        else if (x > tv2) { tv2=x; ti2=v; }
    }

    __shared__ float sm[256], ssum[256], stv[256 * 3];
    __shared__ int   sti[256 * 3];
    const int t = threadIdx.x;
    sm[t] = m; ssum[t] = s;
    stv[t*3+0] = tv0; stv[t*3+1] = tv1; stv[t*3+2] = tv2;
    sti[t*3+0] = ti0; sti[t*3+1] = ti1; sti[t*3+2] = ti2;
    __syncthreads();

    if (t == 0) {
        float M = -INF_F;
        for (int i = 0; i < 256; ++i) M = fmaxf(M, sm[i]);
        float Ssum = 0.f;
        for (int i = 0; i < 256; ++i)
            if (ssum[i] > 0.f) Ssum += ssum[i] * expf(sm[i] - M);
        float bv[3] = { -INF_F, -INF_F, -INF_F };
        int   bi[3] = { 0x7fffffff, 0x7fffffff, 0x7fffffff };
        for (int i = 0; i < 256 * 3; ++i) insert3(stv[i], sti[i], bv, bi);
        const int p = b * NCHUNK + c;
        pm[p] = M; ps[p] = Ssum;
        for (int k = 0; k < 3; ++k) { ptv[p*3+k] = bv[k]; pti[p*3+k] = bi[k]; }
    }
}

// ---- K2: merge partials, beam select, small outputs ------------------------
__global__ void beam_select(const float* __restrict__ pm, const float* __restrict__ ps,
                            const float* __restrict__ ptv, const int* __restrict__ pti,
                            const float* __restrict__ prev, const int* __restrict__ save_id,
                            int* __restrict__ beam, int* __restrict__ tok,
                            float* __restrict__ out) {
    if (threadIdx.x != 0 || blockIdx.x != 0) return;

    float lpv[9]; int lpi[9];
    for (int b = 0; b < 3; ++b) {
        float M = -INF_F;
        for (int c = 0; c < NCHUNK; ++c) M = fmaxf(M, pm[b * NCHUNK + c]);
        float Ssum = 0.f;
        for (int c = 0; c < NCHUNK; ++c) {
            float pmv = pm[b * NCHUNK + c];
            float psv = ps[b * NCHUNK + c];
            if (psv > 0.f) Ssum += psv * expf(pmv - M);
        }
        float bv[3] = { -INF_F, -INF_F, -INF_F };
        int   bi[3] = { 0x7fffffff, 0x7fffffff, 0x7fffffff };
        for (int c = 0; c < NCHUNK * 3; ++c)
            insert3(ptv[b * NCHUNK * 3 + c], pti[b * NCHUNK * 3 + c], bv, bi);
        const float lse = M + logf(Ssum);
        for (int k = 0; k < 3; ++k) { lpv[b*3+k] = bv[k] - lse; lpi[b*3+k] = bi[k]; }
    }

    float cur[9];
    for (int j = 0; j < 9; ++j) cur[j] = lpv[j] + prev[j / 3];

    bool used[9] = {false,false,false,false,false,false,false,false,false};
    for (int sI = 0; sI < 3; ++sI) {
        int bj = -1; float best = -INF_F;
        for (int j = 0; j < 9; ++j)                // ascending scan + strict >
            if (!used[j] && (bj < 0 || cur[j] > best)) { best = cur[j]; bj = j; }
        used[bj] = true;
        beam[sI] = bj / 3;
        tok[sI]  = lpi[bj];
        out[OFF_PROB + sI] = best;
        out[OFF_TOK  + sI] = (float)lpi[bj];
    }
    out[OFF_MAXIDX] = (float)tok[0];
    for (int sI = 0; sI < 3; ++sI) {
        for (int h = 0; h < HIST; ++h)
            out[OFF_SAVE + (size_t)sI * (HIST + 1) + h] =
                (float)save_id[beam[sI] * HIST + h];
        out[OFF_SAVE + (size_t)sI * (HIST + 1) + HIST] = (float)tok[sI];
    }
}

// ---- K3: repeat-penalty gather + single-column multiply --------------------
__global__ __launch_bounds__(256)
void rp_gather(const float* __restrict__ rp, const int* __restrict__ beam,
               const int* __restrict__ tok, const float* __restrict__ pen,
               float* __restrict__ out_rp) {
    const int i = blockIdx.x * 256 + threadIdx.x;
    if (i >= 3 * VOCAB) return;
    const int row = i / VOCAB;
    const int v   = i - row * VOCAB;
    float x = rp[(size_t)beam[row] * VOCAB + v];
    if (v == tok[row]) x *= pen[0];
    out_rp[i] = x;
}

// ---- K4: KV gather via CDNA5 async LDS DMA ---------------------------------
// 84 contiguous 8 MiB segments; 64 blocks/segment; 32 KB LDS staging/round.
// ASYNCcnt is shared by async loads+stores (in-order only within type), so
// phases are separated by full s_wait_asynccnt 0 fences; concurrency comes
// from 8 slots/lane per phase and thousands of resident waves.
static constexpr int BLOCKS_PER_TILE = 64;
static constexpr int SLOTS = 8;                                        // 16B each
static constexpr int CHUNK_BYTES = (int)(TILE_FLOATS * 4 / BLOCKS_PER_TILE); // 131072
static constexpr int ROUND_BYTES = 256 * SLOTS * 16;                   // 32768
static constexpr int ROUNDS = CHUNK_BYTES / ROUND_BYTES;               // 4

__global__ __launch_bounds__(256)
void kv_copy_async(const float* __restrict__ kv, float* __restrict__ out,
                   const int* __restrict__ beam) {
    const int tile = blockIdx.x / BLOCKS_PER_TILE;   // 0..83 == l*3 + b
    const int sub  = blockIdx.x % BLOCKS_PER_TILE;
    const int l = tile / 3;
    const int b = tile % 3;
    const int bi = beam[b];

    const float* src = kv  + ((size_t)(l * 3 + bi)) * TILE_FLOATS
                           + (size_t)sub * (CHUNK_BYTES / 4);
    float*       dst = out + ((size_t)(l * 3 + b )) * TILE_FLOATS
                           + (size_t)sub * (CHUNK_BYTES / 4);

    __shared__ __align__(16) float smem[ROUND_BYTES / 4];    // 32 KB staging
    const unsigned lds_base = (unsigned)(uintptr_t)&smem[0]; // low 32b = LDS offset
    const int tid = threadIdx.x;
    const uint64_t gsrc = (uint64_t)(uintptr_t)src;
    const uint64_t gdst = (uint64_t)(uintptr_t)dst;

    for (int r = 0; r < ROUNDS; ++r) {
#pragma unroll
        for (int k = 0; k < SLOTS; ++k) {
            unsigned lds = lds_base + (unsigned)((tid * SLOTS + k) * 16);
            uint64_t ga  = gsrc + (uint64_t)(((r * SLOTS + k) * 256 + tid) * 16);
            asm volatile("global_load_async_to_lds_b128 %0, %1, off th:TH_LOAD_NT"
                         :: "v"(lds), "v"(ga) : "memory");
        }
        // async LDS writes are unordered vs async LDS reads -> fence on ASYNCcnt
        asm volatile("s_wait_asynccnt 0x0" ::: "memory");
#pragma unroll
        for (int k = 0; k < SLOTS; ++k) {
            unsigned lds = lds_base + (unsigned)((tid * SLOTS + k) * 16);
            uint64_t ga  = gdst + (uint64_t)(((r * SLOTS + k) * 256 + tid) * 16);
            asm volatile("global_store_async_from_lds_b128 %0, %1, off th:TH_STORE_NT"
                         :: "v"(ga), "v"(lds) : "memory");
        }
        // stores must finish reading LDS before next round overwrites it
        asm volatile("s_wait_asynccnt 0x0" ::: "memory");
    }
}

// ---- launcher --------------------------------------------------------------
extern "C" void kernel_launch(void* const* d_in, const int* in_sizes, int n_in,
                              void* d_out, int out_size, void* d_ws, size_t ws_size,
                              hipStream_t stream) {
    const float* kv      = (const float*)d_in[0];
    const float* logits  = (const float*)d_in[1];
    const int*   save_id = (const int*)  d_in[2];
    const float* rp      = (const float*)d_in[3];
    const float* prev    = (const float*)d_in[4];
    const float* pen     = (const float*)d_in[5];
    float* out = (float*)d_out;

    char* ws  = (char*)d_ws;
    int*   beam = (int*)  (ws + WS_BEAM);
    int*   tok  = (int*)  (ws + WS_TOK);
    float* pm   = (float*)(ws + WS_PM);
    float* ps   = (float*)(ws + WS_PS);
    float* ptv  = (float*)(ws + WS_PTV);
    int*   pti  = (int*)  (ws + WS_PTI);

    partial_topk<<<3 * NCHUNK, 256, 0, stream>>>(logits, rp, pm, ps, ptv, pti);
    beam_select<<<1, 32, 0, stream>>>(pm, ps, ptv, pti, prev, save_id, beam, tok, out);
    rp_gather<<<(3 * VOCAB + 255) / 256, 256, 0, stream>>>(rp, beam, tok, pen, out + OFF_RP);
    kv_copy_async<<<(L * B) * BLOCKS_PER_TILE, 256, 0, stream>>>(kv, out + OFF_KV, beam);
}